// BahdanauAttention_63453846831565
// MI455X (gfx1250) — compile-verified
//
#include <hip/hip_runtime.h>
#include <hip/hip_bf16.h>
#include <math.h>

#define B_  8
#define TQ_ 128
#define TK_ 128
#define NH_ 512
#define U_  256
#define NEG_INF_ (-1e9f)

typedef __attribute__((ext_vector_type(2))) float v2f;
typedef __attribute__((ext_vector_type(8))) float v8f;

__device__ __forceinline__ float dev_tanh(float x) {
#if __has_builtin(__builtin_amdgcn_tanhf)
    return __builtin_amdgcn_tanhf(x);   // v_tanh_f32 on gfx1250
#else
    float r;
    asm volatile("v_tanh_f32 %0, %1" : "=v"(r) : "v"(x));
    return r;
#endif
}

// ---------------------------------------------------------------------------
// Projection:  Out = X (rows x NH_) * W (NH_ x U_), f32 WMMA 16x16x4.
// One wave per 16x16 output tile.
//   transpose_out == 0: Out is (B_*T, U_) row-major (q path)
//   transpose_out == 1: Out is (B_, U_/4, TK_, 4) -- u packed in groups of 4
//                       so the score kernel loads one float4 (4 u's) per lane
//                       with fully coalesced 512B/wave accesses (k path).
// ---------------------------------------------------------------------------
__global__ __launch_bounds__(128) void proj_wmma_f32(
    const float* __restrict__ X,
    const float* __restrict__ W,
    float* __restrict__ Out,
    int transpose_out)
{
    const int tilesN = U_ / 16;                              // 16
    int wave = (blockIdx.x * blockDim.x + threadIdx.x) >> 5; // global wave id
    int lane = threadIdx.x & 31;

    int tileM = wave / tilesN;
    int tileN = wave % tilesN;

    int m     = lane & 15;          // row within tile (A) / col within tile (B)
    int khalf = (lane >> 4) * 2;    // which K-pair this half-wave owns

    int arow = tileM * 16 + m;      // global row of X
    int bcol = tileN * 16 + m;      // global col of W

    v8f acc = {};
    #pragma unroll 8
    for (int kk = 0; kk < NH_; kk += 4) {
        // A fragment: 16x4 tile, lane holds A[m][kk+khalf .. +1] (contiguous pair)
        v2f a = *(const v2f*)(X + (size_t)arow * NH_ + kk + khalf);
        // B fragment: 4x16 tile, lane holds B[kk+khalf][n], B[kk+khalf+1][n]
        v2f b;
        b.x = W[(size_t)(kk + khalf)     * U_ + bcol];
        b.y = W[(size_t)(kk + khalf + 1) * U_ + bcol];
        acc = __builtin_amdgcn_wmma_f32_16x16x4_f32(
            /*neg_a=*/false, a, /*neg_b=*/false, b,
            /*c_mod=*/(short)0, acc, /*reuse_a=*/false, /*reuse_b=*/false);
    }

    // D layout: VGPR r -> (M = r + 8*(lane>>4), N = lane&15)
    int outM0 = tileM * 16 + 8 * (lane >> 4);
    int outN  = tileN * 16 + (lane & 15);
    if (!transpose_out) {
        #pragma unroll
        for (int r = 0; r < 8; ++r)
            Out[(size_t)(outM0 + r) * U_ + outN] = acc[r];
    } else {
        #pragma unroll
        for (int r = 0; r < 8; ++r) {
            int gm = outM0 + r;           // gm = b*TK_ + s
            int bb = gm / TK_;
            int ss = gm % TK_;
            int u  = outN;
            Out[(((size_t)bb * (U_ / 4) + (u >> 2)) * TK_ + ss) * 4 + (u & 3)] = acc[r];
        }
    }
}

// ---------------------------------------------------------------------------
// Scores + masked softmax. One 128-thread block per (b, t) row; thread s owns
// score[b,t,s] = sum_u scale[u] * tanh(q[b,t,u] + k[b,s,u]).
// kT4 is (B_, U_/4, TK_, 4): one float4 load covers 4 u's, coalesced over s.
// q and scale reads are wave-uniform (scalar path).
// ---------------------------------------------------------------------------
__global__ __launch_bounds__(TK_) void scores_softmax(
    const float* __restrict__ q,      // (B_, TQ_, U_)
    const float* __restrict__ kT4,    // (B_, U_/4, TK_, 4)
    const float* __restrict__ scale,  // (U_)
    const int*   __restrict__ mask,   // (B_, TK_) nonzero = valid
    float* __restrict__ attn)         // (B_, TQ_, TK_)
{
    int b = blockIdx.x / TQ_;
    int t = blockIdx.x % TQ_;
    int s = threadIdx.x;

    const float4* kb = (const float4*)kT4 + (size_t)b * (U_ / 4) * TK_;
    const float4* q4 = (const float4*)(q + ((size_t)b * TQ_ + t) * U_);
    const float4* s4 = (const float4*)scale;

    float acc = 0.f;
    for (int u0 = 0; u0 < U_ / 4; u0 += 16) {                 // 16 float4 = 64 u per chunk
        if (u0 + 16 < U_ / 4)
            __builtin_prefetch(&kb[(size_t)(u0 + 16) * TK_ + s], 0, 1);
        #pragma unroll
        for (int u4 = u0; u4 < u0 + 16; ++u4) {
            float4 kv = kb[(size_t)u4 * TK_ + s];
            float4 qv = q4[u4];
            float4 sv = s4[u4];
            acc += sv.x * dev_tanh(qv.x + kv.x);
            acc += sv.y * dev_tanh(qv.y + kv.y);
            acc += sv.z * dev_tanh(qv.z + kv.z);
            acc += sv.w * dev_tanh(qv.w + kv.w);
        }
    }

    float sc = (mask[b * TK_ + s] != 0) ? acc : NEG_INF_;

    __shared__ float red[TK_];
    red[s] = sc;
    __syncthreads();
    #pragma unroll
    for (int off = TK_ / 2; off >= 1; off >>= 1) {
        if (s < off) red[s] = fmaxf(red[s], red[s + off]);
        __syncthreads();
    }
    float rowmax = red[0];
    __syncthreads();

    float e = __expf(sc - rowmax);
    red[s] = e;
    __syncthreads();
    #pragma unroll
    for (int off = TK_ / 2; off >= 1; off >>= 1) {
        if (s < off) red[s] += red[s + off];
        __syncthreads();
    }
    float inv = 1.f / red[0];

    attn[((size_t)b * TQ_ + t) * TK_ + s] = e * inv;
}

// ---------------------------------------------------------------------------
// Context: per batch, (TQ_ x TK_) attn * (TK_ x NH_) value, f32 WMMA 16x16x4.
// One wave per 16x16 output tile.
// ---------------------------------------------------------------------------
__global__ __launch_bounds__(128) void context_wmma_f32(
    const float* __restrict__ attn,   // (B_, TQ_, TK_)
    const float* __restrict__ value,  // (B_, TK_, NH_)
    float* __restrict__ ctx)          // (B_, TQ_, NH_)
{
    const int tilesM = TQ_ / 16;  // 8
    const int tilesN = NH_ / 16;  // 32
    int wave = (blockIdx.x * blockDim.x + threadIdx.x) >> 5;
    int lane = threadIdx.x & 31;

    int b    = wave / (tilesM * tilesN);
    int rem  = wave % (tilesM * tilesN);
    int tileM = rem / tilesN;
    int tileN = rem % tilesN;

    int m     = lane & 15;
    int khalf = (lane >> 4) * 2;

    const float* A = attn  + (size_t)b * TQ_ * TK_;
    const float* V = value + (size_t)b * TK_ * NH_;

    v8f acc = {};
    #pragma unroll 8
    for (int ss = 0; ss < TK_; ss += 4) {
        v2f a = *(const v2f*)(A + (size_t)(tileM * 16 + m) * TK_ + ss + khalf);
        v2f bf;
        bf.x = V[(size_t)(ss + khalf)     * NH_ + tileN * 16 + m];
        bf.y = V[(size_t)(ss + khalf + 1) * NH_ + tileN * 16 + m];
        acc = __builtin_amdgcn_wmma_f32_16x16x4_f32(
            false, a, false, bf, (short)0, acc, false, false);
    }

    int outM0 = tileM * 16 + 8 * (lane >> 4);
    int outN  = tileN * 16 + (lane & 15);
    float* C = ctx + (size_t)b * TQ_ * NH_;
    #pragma unroll
    for (int r = 0; r < 8; ++r)
        C[(size_t)(outM0 + r) * NH_ + outN] = acc[r];
}

// ---------------------------------------------------------------------------
extern "C" void kernel_launch(void* const* d_in, const int* in_sizes, int n_in,
                              void* d_out, int out_size, void* d_ws, size_t ws_size,
                              hipStream_t stream) {
    const float* query = (const float*)d_in[0];  // (B_, TQ_, NH_)
    const float* value = (const float*)d_in[1];  // (B_, TK_, NH_)
    const int*   mask  = (const int*)  d_in[2];  // (B_, TK_)
    const float* W1    = (const float*)d_in[3];  // (NH_, U_)
    const float* W2    = (const float*)d_in[4];  // (NH_, U_)
    const float* scale = (const float*)d_in[5];  // (U_)

    float* ctx  = (float*)d_out;                          // (B_, TQ_, NH_)
    float* attn = (float*)d_out + (size_t)B_ * TQ_ * NH_; // (B_, TQ_, TK_)

    float* q   = (float*)d_ws;                            // (B_, TQ_, U_)
    float* kT4 = q + (size_t)B_ * TQ_ * U_;               // (B_, U_/4, TK_, 4)

    // Projections: (B_*128 rows / 16) x (U_ / 16) tiles = 1024 waves = 256 blocks
    {
        int waves  = (B_ * TQ_ / 16) * (U_ / 16);
        int blocks = waves * 32 / 128;
        proj_wmma_f32<<<blocks, 128, 0, stream>>>(query, W1, q,   0);
        proj_wmma_f32<<<blocks, 128, 0, stream>>>(value, W2, kT4, 1);
    }

    // Scores + softmax: one block per (b, t) row
    scores_softmax<<<B_ * TQ_, TK_, 0, stream>>>(q, kT4, scale, mask, attn);

    // Context: 8 * 8 * 32 = 2048 waves = 512 blocks
    {
        int waves  = B_ * (TQ_ / 16) * (NH_ / 16);
        int blocks = waves * 32 / 128;
        context_wmma_f32<<<blocks, 128, 0, stream>>>(attn, value, ctx);
    }
}